// BaseLayer_76055280877648
// MI455X (gfx1250) — compile-verified
//
#include <hip/hip_runtime.h>

#define D_FEAT 32
#define THREADS 256
#define GRID_BLOCKS 4096   // grid-stride; 4096*256/8 = 131072 edges in flight per pass

// ---- scatter kernel first in file so the disasm snippet shows the atomic path ----
__global__ __launch_bounds__(THREADS) void spmm_coo_scatter_kernel(
        const int*   __restrict__ src,   // [E]
        const int*   __restrict__ dst,   // [E]
        const float* __restrict__ attr,  // [E]
        const float* __restrict__ x,     // [N, 32]
        float*       __restrict__ out,   // [N, 32]
        int nE) {
    const int t      = blockIdx.x * blockDim.x + threadIdx.x;
    const int q      = t & 7;                       // float4 slice within the 32-wide row
    const int e0     = t >> 3;                      // first edge for this 8-lane group
    const int stride = (GRID_BLOCKS * THREADS) >> 3;  // edges advanced per grid pass

    for (int e = e0; e < nE; e += stride) {
        // Edge streams are single-use: non-temporal so they don't evict the
        // L2-resident x/out working set. Broadcast within the 8-lane group.
        int   s = __builtin_nontemporal_load(src  + e);
        int   d = __builtin_nontemporal_load(dst  + e);
        float w = __builtin_nontemporal_load(attr + e);

        // One coalesced 128B cacheline per edge row: 8 lanes x global_load_b128.
        const float4* __restrict__ xrow = (const float4*)(x + (size_t)d * D_FEAT);
        float4 v = xrow[q];
        v.x *= w; v.y *= w; v.z *= w; v.w *= w;

        // Non-returning hardware f32 atomics into the L2 atomic units.
        float* __restrict__ o = out + (size_t)s * D_FEAT + (q << 2);
        unsafeAtomicAdd(o + 0, v.x);
        unsafeAtomicAdd(o + 1, v.y);
        unsafeAtomicAdd(o + 2, v.z);
        unsafeAtomicAdd(o + 3, v.w);
    }
}

__global__ __launch_bounds__(THREADS) void zero_f4_kernel(float4* __restrict__ out, int n4) {
    int i = blockIdx.x * blockDim.x + threadIdx.x;
    if (i < n4) out[i] = make_float4(0.f, 0.f, 0.f, 0.f);
}

extern "C" void kernel_launch(void* const* d_in, const int* in_sizes, int n_in,
                              void* d_out, int out_size, void* d_ws, size_t ws_size,
                              hipStream_t stream) {
    // Inputs per reference order: edge_index [2,E] (int), edge_attr [E] (f32), x [N,32] (f32)
    const int*   ei   = (const int*)d_in[0];
    const float* attr = (const float*)d_in[1];
    const float* x    = (const float*)d_in[2];
    float*       out  = (float*)d_out;

    const int nE = in_sizes[1];          // E = edge_attr element count
    const int* src = ei;                 // row 0 of edge_index
    const int* dst = ei + nE;            // row 1 of edge_index

    // Zero-init output (harness poisons d_out with 0xAA).
    int n4 = out_size >> 2;              // out_size = N*32, divisible by 4
    int zb = (n4 + THREADS - 1) / THREADS;
    zero_f4_kernel<<<zb, THREADS, 0, stream>>>((float4*)out, n4);

    // Scatter kernel: 8 threads per edge, grid-stride over edges.
    spmm_coo_scatter_kernel<<<GRID_BLOCKS, THREADS, 0, stream>>>(src, dst, attr, x, out, nE);
}